// CurriculumDynamicThresholdingND_68264210202892
// MI455X (gfx1250) — compile-verified
//
#include <hip/hip_runtime.h>

// ---------------------------------------------------------------------------
// CurriculumDynamicThresholding: B=8, C=19, H=512, W=1024, float32
// out = [ delta (N floats) | T_c (19 floats) | y_hat (N floats) ]
// HBM-bound (319 MB @ 23.3 TB/s ~= 13.7us floor); no matrix ops, so the
// CDNA5 paths used are the async global->LDS DMA (ASYNCcnt) with NT hints.
// ---------------------------------------------------------------------------

#define TAUV 0.6f
#define EPSV 1e-6f

constexpr int BB = 8;
constexpr int CC = 19;
constexpr int HH = 512;
constexpr int WW = 1024;
constexpr int HWQ = HH * WW;        // 524288
constexpr int NPIX = BB * HWQ;      // 4194304

constexpr int T1_THREADS = 128;               // 4 waves (wave32)
constexpr int T1_PIX = T1_THREADS * 4;        // 512 pixels / workgroup
constexpr int T2_THREADS = 256;

typedef float v4f __attribute__((ext_vector_type(4)));

#if __has_builtin(__builtin_amdgcn_global_load_async_to_lds_b128)
#define USE_ASYNC_LDS 1
#else
#define USE_ASYNC_LDS 0
#endif

#if USE_ASYNC_LDS
typedef int v4i __attribute__((ext_vector_type(4)));
typedef __attribute__((address_space(1))) v4i* gptr_v4i;   // global (AS1) int4*
typedef __attribute__((address_space(3))) v4i* lptr_v4i;   // LDS    (AS3) int4*
#endif

static __device__ __forceinline__ void wait_async0() {
#if __has_builtin(__builtin_amdgcn_s_wait_asynccnt)
  __builtin_amdgcn_s_wait_asynccnt(0);
#elif defined(__gfx1250__)
  asm volatile("s_wait_asynccnt 0" ::: "memory");
#endif
}

// ---------------------------------------------------------------------------
// Kernel 0: zero the 19-counter global histogram in workspace
// ---------------------------------------------------------------------------
__global__ void zero_hist_kernel(unsigned* __restrict__ ghist) {
  if (threadIdx.x < CC) ghist[threadIdx.x] = 0u;
}

// ---------------------------------------------------------------------------
// Kernel 1: per-pixel conf / y_hat + global histogram of confident preds.
// conf is parked in the delta slot of d_out (overwritten in pass 2).
// ---------------------------------------------------------------------------
__global__ __launch_bounds__(T1_THREADS)
void pass1_kernel(const float* __restrict__ logits,
                  float* __restrict__ out,
                  unsigned* __restrict__ ghist) {
  __shared__ unsigned hist[4 * 20];           // per-wave sub-histograms (padded)
#if USE_ASYNC_LDS
  __shared__ float stage[CC * T1_PIX];        // 19*512*4 = 38912 B
#endif
  const int tid = threadIdx.x;
  const int pix = blockIdx.x * T1_PIX + tid * 4;     // 16B aligned
  const int b   = pix / HWQ;                          // HWQ multiple of T1_PIX
  const int hw  = pix - b * HWQ;
  const float* gsrc = logits + (size_t)b * CC * HWQ + hw;

#if USE_ASYNC_LDS
  // Kick off all 19 class-row DMAs first: they run while we init the
  // histogram and cross the barrier. cpol=1 -> TH_NT: logits are a
  // stream-once 319MB read; keep L2 free for conf/y_hat reuse in pass 2.
  #pragma unroll
  for (int c = 0; c < CC; ++c) {
    __builtin_amdgcn_global_load_async_to_lds_b128(
        (gptr_v4i)(gsrc + (size_t)c * HWQ),
        (lptr_v4i)(&stage[c * T1_PIX + tid * 4]),
        0, 1);
  }
#endif

  if (tid < 4 * 20) hist[tid] = 0u;
  __syncthreads();

#if USE_ASYNC_LDS
  wait_async0();   // only this thread's own LDS slots are read back
#endif

  float m[4] = {-3.4e38f, -3.4e38f, -3.4e38f, -3.4e38f};
  int   a[4] = {0, 0, 0, 0};

  // Pass A: max + first-argmax over classes
  #pragma unroll
  for (int c = 0; c < CC; ++c) {
#if USE_ASYNC_LDS
    const v4f v = *reinterpret_cast<const v4f*>(&stage[c * T1_PIX + tid * 4]);
#else
    const v4f v = __builtin_nontemporal_load(
        reinterpret_cast<const v4f*>(gsrc + (size_t)c * HWQ));
#endif
    const float x[4] = {v.x, v.y, v.z, v.w};
    #pragma unroll
    for (int k = 0; k < 4; ++k)
      if (x[k] > m[k]) { m[k] = x[k]; a[k] = c; }
  }

  // Pass B: s = sum_c exp(l_c - m)   ->  conf = max softmax = 1/s
  float s[4] = {0.f, 0.f, 0.f, 0.f};
  #pragma unroll
  for (int c = 0; c < CC; ++c) {
#if USE_ASYNC_LDS
    const v4f v = *reinterpret_cast<const v4f*>(&stage[c * T1_PIX + tid * 4]);
#else
    const v4f v = __builtin_nontemporal_load(
        reinterpret_cast<const v4f*>(gsrc + (size_t)c * HWQ));
#endif
    s[0] += __expf(v.x - m[0]);
    s[1] += __expf(v.y - m[1]);
    s[2] += __expf(v.z - m[2]);
    s[3] += __expf(v.w - m[3]);
  }

  const int wv = tid >> 5;
  v4f cf;
  cf.x = 1.0f / s[0]; cf.y = 1.0f / s[1]; cf.z = 1.0f / s[2]; cf.w = 1.0f / s[3];
  *reinterpret_cast<v4f*>(&out[pix]) = cf;           // conf -> delta slot (temp)

  const float cfa[4] = {cf.x, cf.y, cf.z, cf.w};
  #pragma unroll
  for (int k = 0; k < 4; ++k) {
    out[(size_t)NPIX + CC + pix + k] = (float)a[k];  // y_hat slot (scalar: +19 skew)
    if (cfa[k] > TAUV) atomicAdd(&hist[wv * 20 + a[k]], 1u);
  }

  __syncthreads();
  if (tid < CC) {
    unsigned t = hist[tid] + hist[20 + tid] + hist[40 + tid] + hist[60 + tid];
    if (t) atomicAdd(&ghist[tid], t);
  }
}

// ---------------------------------------------------------------------------
// Kernel 2: T_c from histogram (single wave32, shuffle max-reduction)
// ---------------------------------------------------------------------------
__global__ void tc_kernel(const unsigned* __restrict__ ghist,
                          float* __restrict__ out) {
  const int t = threadIdx.x;
  const float sig = (t < CC) ? (float)ghist[t] : 0.0f;
  float mx = sig;
  #pragma unroll
  for (int off = 16; off > 0; off >>= 1)
    mx = fmaxf(mx, __shfl_xor(mx, off, 32));
  const float denom = fmaxf(mx, EPSV);
  const float sh = sig / denom;
  const float tc = sh / (2.0f - fminf(sh, 1.0f)) * TAUV;
  if (t < CC) out[(size_t)NPIX + t] = tc;
}

// ---------------------------------------------------------------------------
// Kernel 3: delta = conf > T_c[y_hat], in place over the conf slot
// ---------------------------------------------------------------------------
__global__ __launch_bounds__(T2_THREADS)
void pass2_kernel(float* __restrict__ out) {
  __shared__ float Ts[CC];
  const int tid = threadIdx.x;
  if (tid < CC) Ts[tid] = out[(size_t)NPIX + tid];
  __syncthreads();

  const int pix = (blockIdx.x * T2_THREADS + tid) * 4;
  const v4f cf = *reinterpret_cast<const v4f*>(&out[pix]);   // L2-resident
  int y[4];
  #pragma unroll
  for (int k = 0; k < 4; ++k)
    y[k] = (int)out[(size_t)NPIX + CC + pix + k];            // L2-resident

  v4f d;
  d.x = (cf.x > Ts[y[0]]) ? 1.0f : 0.0f;
  d.y = (cf.y > Ts[y[1]]) ? 1.0f : 0.0f;
  d.z = (cf.z > Ts[y[2]]) ? 1.0f : 0.0f;
  d.w = (cf.w > Ts[y[3]]) ? 1.0f : 0.0f;
  __builtin_nontemporal_store(d, reinterpret_cast<v4f*>(&out[pix]));
}

// ---------------------------------------------------------------------------
extern "C" void kernel_launch(void* const* d_in, const int* in_sizes, int n_in,
                              void* d_out, int out_size, void* d_ws, size_t ws_size,
                              hipStream_t stream) {
  (void)in_sizes; (void)n_in; (void)out_size; (void)ws_size;
  const float* logits = (const float*)d_in[0];
  float* out = (float*)d_out;
  unsigned* ghist = (unsigned*)d_ws;   // 19 * 4 bytes of scratch

  zero_hist_kernel<<<1, 32, 0, stream>>>(ghist);
  pass1_kernel<<<NPIX / T1_PIX, T1_THREADS, 0, stream>>>(logits, out, ghist);
  tc_kernel<<<1, 32, 0, stream>>>(ghist, out);
  pass2_kernel<<<NPIX / (T2_THREADS * 4), T2_THREADS, 0, stream>>>(out);
}